// EdgeModel_4750233829497
// MI455X (gfx1250) — compile-verified
//
#include <hip/hip_runtime.h>

typedef float v2f __attribute__((ext_vector_type(2)));
typedef float v4f __attribute__((ext_vector_type(4)));
typedef float v8f __attribute__((ext_vector_type(8)));

__device__ __forceinline__ v8f wmma4(v2f a, v2f b, v8f c) {
  // V_WMMA_F32_16X16X4_F32 : D = A(16x4) * B(4x16) + C(16x16)
  return __builtin_amdgcn_wmma_f32_16x16x4_f32(false, a, false, b, (short)0, c,
                                               false, false);
}

// butterfly add within groups of 16 lanes via ds_swizzle (group-of-32 XOR mode)
__device__ __forceinline__ float red16(float v) {
  v += __int_as_float(
      __builtin_amdgcn_ds_swizzle(__float_as_int(v), (1 << 10) | 0x1f));
  v += __int_as_float(
      __builtin_amdgcn_ds_swizzle(__float_as_int(v), (2 << 10) | 0x1f));
  v += __int_as_float(
      __builtin_amdgcn_ds_swizzle(__float_as_int(v), (4 << 10) | 0x1f));
  v += __int_as_float(
      __builtin_amdgcn_ds_swizzle(__float_as_int(v), (8 << 10) | 0x1f));
  return v;
}

__global__ void __launch_bounds__(256) zero_f32(float* __restrict__ p, int n) {
  int i = blockIdx.x * blockDim.x + threadIdx.x;
  if (i < n) p[i] = 0.f;
}

// ---------------------------------------------------------------------------
// Fused NNConv edge pass. One wave handles 16 edges.
//   hid = relu(edge_attr_tile @ W1 + b1)                 (2-step WMMA, K=8)
//   msg[m,o] = Sum_{i,k} nf[src_m,i]*hid[m,k]*W2[k,i*OSTR+o]
//            + Sum_i    nf[src_m,i]*b2[i*OSTR+o]
// as ONE WMMA chain with A = per-edge outer product, K mapped k-major
// (K = k*NI + i, then bias rows K = 10*NI + i). Since NI % 4 == 0,
// K mod 4 == i mod 4: each lane's node-feature operands are a contiguous
// b64 at byte offset 8*hv -> loaded directly, no selects. Each A element
// is then a single v_mul (nsv * hr[k]).
// ---------------------------------------------------------------------------
template <int NI, int WCOLS, int OSTR, int OW>
__global__ void __launch_bounds__(256)
edge_conv(const float* __restrict__ nfeat, const float* __restrict__ eattr,
          const int* __restrict__ eidx, const float* __restrict__ w1,
          const float* __restrict__ b1, const float* __restrict__ w2,
          const float* __restrict__ b2, float* __restrict__ agg, int E_,
          int ntiles) {
  constexpr int NU = NI / 4;           // A K-steps per k
  constexpr int KOUT = NI * 10;        // outer-product K extent
  constexpr int NSTEP = 10 * NU + NU;  // + bias rows
  __shared__ float sHid[8][16][16];
  __shared__ int sDst[8][16];
  const int lane = threadIdx.x & 31;
  const int wib = threadIdx.x >> 5;
  const int q = lane & 15;   // column (B/C) or row (A) index
  const int hv = lane >> 4;  // half-wave selector
  const int wgid = blockIdx.x * 8 + wib;
  const int nw = gridDim.x * 8;

  // --- hoisted B fragments (uniform weights, resident across tile loop) ---
  v2f bf1[2];
#pragma unroll
  for (int s = 0; s < 2; ++s)
#pragma unroll
    for (int j = 0; j < 2; ++j) {
      int K = 4 * s + 2 * hv + j;
      bf1[s][j] = (q < 10) ? w1[K * 10 + q] : 0.f;
    }
  const float b1v = (q < 10) ? b1[q] : 0.f;

  v2f bf2[NSTEP];
#pragma unroll
  for (int s = 0; s < NSTEP; ++s)
#pragma unroll
    for (int j = 0; j < 2; ++j) {
      int K = 4 * s + 2 * hv + j;
      float v;
      if (K < KOUT) {
        int k = K / NI, i = K % NI;  // NI power of two -> shifts
        v = (q < OW) ? w2[k * WCOLS + i * OSTR + q] : 0.f;
      } else {
        int i = K - KOUT;
        v = (q < OW) ? b2[i * OSTR + q] : 0.f;
      }
      bf2[s][j] = v;
    }

  float* wl = &sHid[wib][0][0];

  for (int t = wgid; t < ntiles; t += nw) {
    const int e0 = t * 16;
    const int eS = eidx[e0 + q];       // src of edge e0+q
    const int eD = eidx[E_ + e0 + q];  // dst of edge e0+q

    // ---- GEMM1: hid = relu(ea_tile @ W1 + b1) ----
    v8f hid;
#pragma unroll
    for (int r = 0; r < 8; ++r) hid[r] = b1v;
    const float* ea = eattr + (size_t)(e0 + q) * 8 + 2 * hv;
    v2f a0 = *(const v2f*)(ea);
    v2f a1 = *(const v2f*)(ea + 4);
    hid = wmma4(a0, bf1[0], hid);
    hid = wmma4(a1, bf1[1], hid);
#pragma unroll
    for (int r = 0; r < 8; ++r) hid[r] = fmaxf(hid[r], 0.f);

    // ---- transpose hid to lane-major + stage dst indices via LDS ----
#pragma unroll
    for (int r = 0; r < 8; ++r) wl[(r + 8 * hv) * 16 + q] = hid[r];
    if (lane < 16) sDst[wib][q] = eD;
    __asm__ volatile("s_wait_dscnt 0" ::: "memory");

    float hr[12];
    {
      v4f t0 = *(const v4f*)&wl[q * 16 + 0];
      v4f t1 = *(const v4f*)&wl[q * 16 + 4];
      v4f t2 = *(const v4f*)&wl[q * 16 + 8];
      hr[0] = t0.x; hr[1] = t0.y; hr[2] = t0.z; hr[3] = t0.w;
      hr[4] = t1.x; hr[5] = t1.y; hr[6] = t1.z; hr[7] = t1.w;
      hr[8] = t2.x; hr[9] = t2.y; hr[10] = t2.z; hr[11] = t2.w;
    }
    // per-lane node-feature operand pairs: i = 4u + 2*hv + {0,1}
    v2f nsv[NU];
#pragma unroll
    for (int u = 0; u < NU; ++u)
      nsv[u] = *(const v2f*)(nfeat + (size_t)eS * NI + 4 * u + 2 * hv);

    int dstm[8];
#pragma unroll
    for (int r = 0; r < 8; ++r) dstm[r] = sDst[wib][r + 8 * hv];

    // ---- fused weight-gen x matvec: msg = (nf ⊗ hid | nf) @ B' ----
    v8f msg;
#pragma unroll
    for (int r = 0; r < 8; ++r) msg[r] = 0.f;
#pragma unroll
    for (int k = 0; k < 10; ++k)
#pragma unroll
      for (int u = 0; u < NU; ++u) {
        v2f a;
        a[0] = nsv[u].x * hr[k];
        a[1] = nsv[u].y * hr[k];
        msg = wmma4(a, bf2[k * NU + u], msg);
      }
#pragma unroll
    for (int u = 0; u < NU; ++u)  // bias rows: A element = nf directly
      msg = wmma4(nsv[u], bf2[10 * NU + u], msg);

    // ---- scatter-add (C-layout: lane q = channel, row r = edge) ----
#pragma unroll
    for (int r = 0; r < 8; ++r) {
      if (OW == 16 || q < OW)
        atomicAdd(&agg[(size_t)dstm[r] * 16 + q], msg[r]);
    }
  }
}

// h0 = relu(agg0 + x @ root + bias), [N,16]
__global__ void __launch_bounds__(256)
node_relu0(const float* __restrict__ x, const float* __restrict__ agg,
           const float* __restrict__ root, const float* __restrict__ bias,
           float* __restrict__ h, int N_) {
  int idx = blockIdx.x * blockDim.x + threadIdx.x;
  if (idx >= N_ * 16) return;
  int v = idx >> 4, j = idx & 15;
  float acc = agg[idx] + bias[j];
#pragma unroll
  for (int i = 0; i < 8; ++i) acc += x[v * 8 + i] * root[i * 16 + j];
  h[idx] = fmaxf(acc, 0.f);
}

// h1 = relu(agg1 + h0 @ root + bias), padded [N,16] (cols 10..15 = 0)
__global__ void __launch_bounds__(256)
node_relu1(const float* __restrict__ h0, const float* __restrict__ agg,
           const float* __restrict__ root, const float* __restrict__ bias,
           float* __restrict__ h1, int N_) {
  int idx = blockIdx.x * blockDim.x + threadIdx.x;
  if (idx >= N_ * 16) return;
  int v = idx >> 4, j = idx & 15;
  float out = 0.f;
  if (j < 10) {
    float acc = agg[idx] + bias[j];
#pragma unroll
    for (int i = 0; i < 16; ++i) acc += h0[v * 16 + i] * root[i * 10 + j];
    out = fmaxf(acc, 0.f);
  }
  h1[idx] = out;
}

// scores = relu([ea | h1[src]] @ W1 + b1) @ w2 + b2 ; one wave per 16 edges
__global__ void __launch_bounds__(256)
edge_pred(const float* __restrict__ eattr, const int* __restrict__ eidx,
          const float* __restrict__ h1, const float* __restrict__ w1,
          const float* __restrict__ b1, const float* __restrict__ w2,
          const float* __restrict__ b2, float* __restrict__ out, int E_,
          int ntiles) {
  const int lane = threadIdx.x & 31;
  const int q = lane & 15;
  const int hv = lane >> 4;
  const int wgid = blockIdx.x * 8 + (threadIdx.x >> 5);
  const int nw = gridDim.x * 8;
  (void)E_;

  v2f bf[5];
#pragma unroll
  for (int s = 0; s < 5; ++s)
#pragma unroll
    for (int j = 0; j < 2; ++j) {
      int K = 4 * s + 2 * hv + j;
      bf[s][j] = (K < 18 && q < 10) ? w1[K * 10 + q] : 0.f;
    }
  const float b1v = (q < 10) ? b1[q] : 0.f;
  const float w2q = (q < 10) ? w2[q] : 0.f;
  const float b2v = b2[0];

  for (int t = wgid; t < ntiles; t += nw) {
    const int e0 = t * 16;
    const int eS = eidx[e0 + q];
    // A fragments loaded directly at half-wave offset (K = 4s + 2hv + j).
    // s=4, hv=1 needs K=18,19 == 0 -> reads h1 cols 10,11 which are zero.
    const float* ear = eattr + (size_t)(e0 + q) * 8 + 2 * hv;
    const float* h1r = h1 + (size_t)eS * 16 + 2 * hv;
    v2f af[5];
    af[0] = *(const v2f*)(ear);
    af[1] = *(const v2f*)(ear + 4);
    af[2] = *(const v2f*)(h1r);
    af[3] = *(const v2f*)(h1r + 4);
    af[4] = *(const v2f*)(h1r + 8);

    v8f d;
#pragma unroll
    for (int r = 0; r < 8; ++r) d[r] = b1v;
#pragma unroll
    for (int s = 0; s < 5; ++s) d = wmma4(af[s], bf[s], d);

    // reduce over the 16 output channels; lane q==r keeps edge (8*hv + r)
    float sc = 0.f;
#pragma unroll
    for (int r = 0; r < 8; ++r) {
      float val = red16(fmaxf(d[r], 0.f) * w2q);
      sc = (q == r) ? (val + b2v) : sc;
    }
    if (q < 8) out[e0 + 8 * hv + q] = sc;
  }
}

extern "C" void kernel_launch(void* const* d_in, const int* in_sizes, int n_in,
                              void* d_out, int out_size, void* d_ws,
                              size_t ws_size, hipStream_t stream) {
  const float* x = (const float*)d_in[0];
  const float* eattr = (const float*)d_in[1];
  const int* eidx = (const int*)d_in[2];
  const float* c0_w1 = (const float*)d_in[3];
  const float* c0_b1 = (const float*)d_in[4];
  const float* c0_w2 = (const float*)d_in[5];
  const float* c0_b2 = (const float*)d_in[6];
  const float* c0_root = (const float*)d_in[7];
  const float* c0_bias = (const float*)d_in[8];
  const float* c1_w1 = (const float*)d_in[9];
  const float* c1_b1 = (const float*)d_in[10];
  const float* c1_w2 = (const float*)d_in[11];
  const float* c1_b2 = (const float*)d_in[12];
  const float* c1_root = (const float*)d_in[13];
  const float* c1_bias = (const float*)d_in[14];
  const float* ep_w1 = (const float*)d_in[15];
  const float* ep_b1 = (const float*)d_in[16];
  const float* ep_w2 = (const float*)d_in[17];
  const float* ep_b2 = (const float*)d_in[18];
  (void)n_in; (void)out_size; (void)ws_size;

  const int N_ = in_sizes[0] / 8;
  const int E_ = in_sizes[1] / 8;
  const int ntiles = E_ / 16;

  float* agg0 = (float*)d_ws;
  float* agg1 = agg0 + (size_t)N_ * 16;
  float* h0 = agg1 + (size_t)N_ * 16;
  float* h1 = h0 + (size_t)N_ * 16;

  const int EBLK = 960;  // persistent waves, grid-stride over edge tiles
  const int nthr = 256;
  const int nblkN = (N_ * 16 + nthr - 1) / nthr;

  zero_f32<<<(2 * N_ * 16 + nthr - 1) / nthr, nthr, 0, stream>>>(agg0,
                                                                 2 * N_ * 16);
  edge_conv<8, 128, 16, 16><<<EBLK, nthr, 0, stream>>>(
      x, eattr, eidx, c0_w1, c0_b1, c0_w2, c0_b2, agg0, E_, ntiles);
  node_relu0<<<nblkN, nthr, 0, stream>>>(x, agg0, c0_root, c0_bias, h0, N_);
  edge_conv<16, 160, 10, 10><<<EBLK, nthr, 0, stream>>>(
      h0, eattr, eidx, c1_w1, c1_b1, c1_w2, c1_b2, agg1, E_, ntiles);
  node_relu1<<<nblkN, nthr, 0, stream>>>(h0, agg1, c1_root, c1_bias, h1, N_);
  edge_pred<<<EBLK, nthr, 0, stream>>>(eattr, eidx, h1, ep_w1, ep_b1, ep_w2,
                                       ep_b2, (float*)d_out, E_, ntiles);
}